// DozatManning_51238959841932
// MI455X (gfx1250) — compile-verified
//
#include <hip/hip_runtime.h>

// ---------------------------------------------------------------------------
// Dozat-Manning biaffine parser forward pass for MI455X (gfx1250, wave32).
// All GEMMs run through V_WMMA_F32_16X16X4_F32 (exact f32, matches reference).
// bT=1 (weights (N,K)) GEMMs: workgroup-cooperative 64x32 macro-tiles, K
// staged through LDS in 64-wide chunks via GLOBAL_LOAD_ASYNC_TO_LDS_B128
// (ASYNCcnt) when available, else explicit b128 load/store; fragments read
// with conflict-free b64 LDS loads. bT=0 GEMMs: branch-free wave-per-tile.
// All K % 4 == 0 by construction.
// ---------------------------------------------------------------------------

typedef float v2f __attribute__((ext_vector_type(2)));
typedef float v8f __attribute__((ext_vector_type(8)));
typedef int   v4i __attribute__((vector_size(16)));   // matches builtin param

#define AS1 __attribute__((address_space(1)))
#define AS3 __attribute__((address_space(3)))

#if __has_builtin(__builtin_amdgcn_global_load_async_to_lds_b128)
#define HAVE_ASYNC_LDS 1
#else
#define HAVE_ASYNC_LDS 0
#endif

__device__ __forceinline__ void wait_async_then_barrier() {
#if HAVE_ASYNC_LDS
#if __has_builtin(__builtin_amdgcn_s_wait_asynccnt)
  __builtin_amdgcn_s_wait_asynccnt(0);
#else
  asm volatile("s_wait_asynccnt 0" ::: "memory");
#endif
#endif
  __syncthreads();
}

static constexpr int Bc   = 128;   // batch
static constexpr int Lc   = 100;   // seq len
static constexpr int HIDc = 400;
static constexpr int GDc  = 4 * HIDc;  // 1600 gate dim
static constexpr int ROWS = Bc * Lc;   // 12800
static constexpr int ARCc = 500;
static constexpr int RELc = 100;
static constexpr int RELP = 104;   // padded rel width (101 -> 104, K%4==0)
static constexpr int LBLc = 50;
static constexpr int LDSP = 68;    // LDS row stride (floats): bank-conflict-free

// C[b][m][n] = act( (Ci ? Ci[b][m][n] : 0) + sum_k A[b][m][k]*Bm[b][k/n] + bias )
struct GemmArgs {
  const float* A;  long long sAb;  int rsA;
  const float* Bm; long long sBb;  int rsB;
  const float* Ci; long long sCib; int rsCi;
  float*       C;  long long sCb;  int rsC; int csC;
  const float* b1; long long sB1;            // per-batch-strided bias (W_rel rows)
  const float* b2;                           // shared bias
  int M, N, K, batch, act;                   // act: 0=none, 1=leaky_relu(0.1)
};

// one 16B global->LDS move (async when the toolchain exposes the builtin)
__device__ __forceinline__ void stage16(const float* gsrc, float* ldst) {
#if HAVE_ASYNC_LDS
  __builtin_amdgcn_global_load_async_to_lds_b128(
      (AS1 v4i*)gsrc, (AS3 v4i*)ldst, 0, 0);
#else
  *(float4*)ldst = *(const float4*)gsrc;
#endif
}

// ---------------- bT=1: LDS-staged cooperative macro-tile GEMM --------------
// Bm is (N,K). Requires: K%4==0; A/B row+batch strides and bases 16B-aligned.
__global__ void wmma_gemm_bt1_lds(GemmArgs ga) {
  __shared__ float As[64 * LDSP];
  __shared__ float Bs[32 * LDSP];
  const int tid  = threadIdx.x;
  const int lane = tid & 31;
  const int wave = tid >> 5;
  const int l15  = lane & 15;
  const int half = lane >> 4;
  const int kh   = half << 1;          // lane's k sub-offset within a 4-step
  const int mw   = (wave >> 1) << 4;   // wave's M sub-tile: 0,16,32,48
  const int nw   = (wave &  1) << 4;   // wave's N sub-tile: 0,16

  const int macroM = (ga.M + 63) >> 6;
  const int macroN = (ga.N + 31) >> 5;
  const long long total = (long long)ga.batch * macroM * macroN;

  for (long long tile = blockIdx.x; tile < total; tile += gridDim.x) {
    const int tn = (int)(tile % macroN);
    const long long rest = tile / macroN;
    const int tm = (int)(rest % macroM);
    const int bb = (int)(rest / macroM);

    const float* Ap = ga.A  + (long long)bb * ga.sAb;
    const float* Bp = ga.Bm + (long long)bb * ga.sBb;

    const int m_base = tm << 6, n_base = tn << 5;
    const int cm0 = m_base + mw, cn0 = n_base + nw;  // wave's 16x16 C tile
    const int bn  = min(cn0 + l15, ga.N - 1);        // clamped C/B column

    v8f acc;
    if (ga.Ci) {
      const float* Cip = ga.Ci + (long long)bb * ga.sCib;
#pragma unroll
      for (int r = 0; r < 8; ++r) {
        const int cm = min(cm0 + (half << 3) + r, ga.M - 1);
        acc[r] = Cip[(long long)cm * ga.rsCi + bn];
      }
    } else {
#pragma unroll
      for (int r = 0; r < 8; ++r) acc[r] = 0.f;
    }

    for (int k0 = 0; k0 < ga.K; k0 += 64) {
      const int kc = min(64, ga.K - k0);   // chunk width, always multiple of 4
      __syncthreads();                     // LDS free of prior-chunk readers
      // stage A: 64 rows x kc floats, coalesced 16B moves (rows clamped).
      // Columns >= kc are never read by the compute loop -> no zero-fill.
#pragma unroll
      for (int it = 0; it < 4; ++it) {
        const int slot = tid + (it << 8);            // 0..1023 float4 slots
        const int row = slot >> 4, c4 = (slot & 15) << 2;
        const int gm = min(m_base + row, ga.M - 1);
        if (c4 < kc)
          stage16(Ap + (long long)gm * ga.rsA + k0 + c4, &As[row * LDSP + c4]);
      }
      // stage B: 32 rows x kc floats
#pragma unroll
      for (int it = 0; it < 2; ++it) {
        const int slot = tid + (it << 8);            // 0..511 float4 slots
        const int row = slot >> 4, c4 = (slot & 15) << 2;
        const int gn = min(n_base + row, ga.N - 1);
        if (c4 < kc)
          stage16(Bp + (long long)gn * ga.rsB + k0 + c4, &Bs[row * LDSP + c4]);
      }
      wait_async_then_barrier();   // drain own ASYNCcnt, then workgroup barrier
      // fragments from LDS: a[r]=A[m][kk+2*half+r], b mirrors; b64, no conflicts
      const float* Asw = &As[(mw + l15) * LDSP + kh];
      const float* Bsw = &Bs[(nw + l15) * LDSP + kh];
#pragma unroll 4
      for (int kk = 0; kk < kc; kk += 4) {
        v2f a = *(const v2f*)(Asw + kk);
        v2f b = *(const v2f*)(Bsw + kk);
        acc = __builtin_amdgcn_wmma_f32_16x16x4_f32(false, a, false, b,
                                                    (short)0, acc, false, false);
      }
    }

    float bias = 0.f;
    if (ga.b1) bias += (ga.b1 + (long long)bb * ga.sB1)[bn];
    if (ga.b2) bias += ga.b2[bn];
    float* Cp = ga.C + (long long)bb * ga.sCb;
    const bool n_ok = (cn0 + l15) < ga.N;
#pragma unroll
    for (int r = 0; r < 8; ++r) {
      const int cm = cm0 + (half << 3) + r;
      if (cm < ga.M && n_ok) {
        float v = acc[r] + bias;
        if (ga.act == 1) v = v > 0.f ? v : 0.1f * v;
        Cp[(long long)cm * ga.rsC + (long long)bn * ga.csC] = v;
      }
    }
  }
}

// ---------------- bT=0: wave-per-tile GEMM, Bm is (K,N) ---------------------
__global__ void wmma_gemm_bt0(GemmArgs ga) {
  const int lane = threadIdx.x & 31;
  const int wave = threadIdx.x >> 5;
  const int wpb  = blockDim.x >> 5;
  const int l15  = lane & 15;
  const int half = lane >> 4;
  const int kh   = half << 1;

  const int tilesM = (ga.M + 15) >> 4;
  const int tilesN = (ga.N + 15) >> 4;
  const long long total = (long long)ga.batch * tilesM * tilesN;

  for (long long tile = (long long)blockIdx.x * wpb + wave; tile < total;
       tile += (long long)gridDim.x * wpb) {
    const int tn = (int)(tile % tilesN);
    const long long rest = tile / tilesN;
    const int tm = (int)(rest % tilesM);
    const int bb = (int)(rest / tilesM);

    const float* Ap = ga.A  + (long long)bb * ga.sAb;
    const float* Bp = ga.Bm + (long long)bb * ga.sBb;

    const int m0 = tm << 4, n0 = tn << 4;
    const int am = min(m0 + l15, ga.M - 1);
    const int bn = min(n0 + l15, ga.N - 1);
    const float* Arow = Ap + (long long)am * ga.rsA;
    const float* Bcol = Bp + bn;

    v8f acc;
#pragma unroll
    for (int r = 0; r < 8; ++r) acc[r] = 0.f;

#pragma unroll 4
    for (int k0 = 0; k0 < ga.K; k0 += 4) {
      v2f a = *(const v2f*)(Arow + k0 + kh);
      v2f b;
      b.x = Bcol[(long long)(k0 + kh    ) * ga.rsB];
      b.y = Bcol[(long long)(k0 + kh + 1) * ga.rsB];
      acc = __builtin_amdgcn_wmma_f32_16x16x4_f32(false, a, false, b,
                                                  (short)0, acc, false, false);
    }

    float bias = 0.f;
    if (ga.b1) bias += (ga.b1 + (long long)bb * ga.sB1)[bn];
    if (ga.b2) bias += ga.b2[bn];
    float* Cp = ga.C + (long long)bb * ga.sCb;
    const bool n_ok = (n0 + l15) < ga.N;
#pragma unroll
    for (int r = 0; r < 8; ++r) {
      const int cm = m0 + (half << 3) + r;
      if (cm < ga.M && n_ok) {
        float v = acc[r] + bias;
        if (ga.act == 1) v = v > 0.f ? v : 0.1f * v;
        Cp[(long long)cm * ga.rsC + (long long)bn * ga.csC] = v;
      }
    }
  }
}

// ------------------------------ helpers ------------------------------------
__device__ __forceinline__ float sigmoidf_(float x) {
  return 1.f / (1.f + __expf(-x));
}

__global__ void zero_kernel(float* p, int n) {
  int i = blockIdx.x * blockDim.x + threadIdx.x;
  if (i < n) p[i] = 0.f;
}

// x[(t*B + b)*200 + k] = concat(word_embed[words[b,t]], feat_embed[feats[b,t]])
__global__ void embed_kernel(const int* __restrict__ words, const int* __restrict__ feats,
                             const float* __restrict__ wemb, const float* __restrict__ femb,
                             float* __restrict__ x) {
  const int din = 200;
  long long i = (long long)blockIdx.x * blockDim.x + threadIdx.x;
  if (i >= (long long)ROWS * din) return;
  int k = (int)(i % din);
  int row = (int)(i / din);           // row = t*B + b
  int b = row % Bc, t = row / Bc;
  float v;
  if (k < 100) v = wemb[(long long)words[b * Lc + t] * 100 + k];
  else         v = femb[(long long)feats[b * Lc + t] * 100 + (k - 100)];
  x[(long long)row * din + k] = v;
}

// gates -> (h,c) update with mask; emits masked h into x_out[t]
__global__ void lstm_cell_kernel(const float* __restrict__ g, const int* __restrict__ words,
                                 float* __restrict__ h, float* __restrict__ c,
                                 float* __restrict__ xout, int t, int colOff, int outStride) {
  int i = blockIdx.x * blockDim.x + threadIdx.x;
  if (i >= Bc * HIDc) return;
  int b = i / HIDc, j = i % HIDc;
  const float* gb = g + (long long)b * GDc;
  float ig = sigmoidf_(gb[j]);
  float fg = sigmoidf_(gb[HIDc + j]);
  float gg = tanhf(gb[2 * HIDc + j]);
  float og = sigmoidf_(gb[3 * HIDc + j]);
  float cn = fg * c[i] + ig * gg;
  float hn = og * tanhf(cn);
  bool m = words[b * Lc + t] != 0;
  if (m) { c[i] = cn; h[i] = hn; }
  xout[(long long)t * Bc * outStride + (long long)b * outStride + colOff + j] = m ? hn : 0.f;
}

__global__ void set_ones_col(float* p, int rows, int stride, int col) {
  int i = blockIdx.x * blockDim.x + threadIdx.x;
  if (i < rows) p[(long long)i * stride + col] = 1.f;
}

// s_arc[b,x,y] = -inf where words[b,y]==PAD
__global__ void arc_mask_kernel(float* s_arc, const int* __restrict__ words) {
  int i = blockIdx.x * blockDim.x + threadIdx.x;
  if (i >= Bc * Lc * Lc) return;
  int y = i % Lc;
  int b = i / (Lc * Lc);
  if (words[b * Lc + y] == 0) s_arc[i] = -__builtin_inff();
}

// ------------------------------ host side ----------------------------------
static inline void run_gemm(hipStream_t st, const GemmArgs& ga, int bT) {
  if (bT) {
    long long mM = (ga.M + 63) / 64, mN = (ga.N + 31) / 32;
    long long tiles = (long long)ga.batch * mM * mN;
    long long blocks = tiles > 16384 ? 16384 : (tiles < 1 ? 1 : tiles);
    wmma_gemm_bt1_lds<<<dim3((unsigned)blocks), dim3(256), 0, st>>>(ga);
  } else {
    long long tilesM = (ga.M + 15) / 16, tilesN = (ga.N + 15) / 16;
    long long tiles = (long long)ga.batch * tilesM * tilesN;
    long long blocks = (tiles + 7) / 8;               // 8 waves (tiles) per WG
    if (blocks > 32768) blocks = 32768;
    if (blocks < 1) blocks = 1;
    wmma_gemm_bt0<<<dim3((unsigned)blocks), dim3(256), 0, st>>>(ga);
  }
}

extern "C" void kernel_launch(void* const* d_in, const int* in_sizes, int n_in,
                              void* d_out, int out_size, void* d_ws, size_t ws_size,
                              hipStream_t stream) {
  (void)in_sizes; (void)n_in; (void)out_size; (void)ws_size;
  auto F = [&](int i) { return (const float*)d_in[i]; };
  const int* words = (const int*)d_in[0];
  const int* feats = (const int*)d_in[1];
  int p = 2;
  const float* wemb = F(p++);
  const float* femb = F(p++);
  const float *Wih[3][2], *Whh[3][2], *bih[3][2], *bhh[3][2];
  for (int l = 0; l < 3; ++l)
    for (int d = 0; d < 2; ++d) {       // d=0 forward ('f'), d=1 backward ('b')
      Wih[l][d] = F(p++); Whh[l][d] = F(p++); bih[l][d] = F(p++); bhh[l][d] = F(p++);
    }
  const float* arcHW = F(p++); const float* arcHb = F(p++);
  const float* arcDW = F(p++); const float* arcDb = F(p++);
  const float* relHW = F(p++); const float* relHb = F(p++);
  const float* relDW = F(p++); const float* relDb = F(p++);
  const float* Warc  = F(p++);   // (501, 500)
  const float* Wrel  = F(p++);   // (50, 101, 101)

  float* out = (float*)d_out;            // [s_arc 1.28M | s_rel 64M]
  float* s_rel = out + Bc * Lc * Lc;

  // workspace carve-up (floats) — ~258 MB total, all offsets multiples of 4
  float* ws    = (float*)d_ws;
  float* xA    = ws;                    // 12800*800
  float* xB    = xA + ROWS * 800;       // 12800*800
  float* XW    = xB + ROWS * 800;       // 12800*1600
  float* gbuf  = XW + (long long)ROWS * GDc;   // 128*1600
  float* hbuf  = gbuf + Bc * GDc;       // 128*400
  float* cbuf  = hbuf + Bc * HIDc;      // 128*400
  float* arcHo = cbuf + Bc * HIDc;      // 12800*500
  float* arcDo = arcHo + ROWS * ARCc;   // 12800*500
  float* Ubuf  = arcDo + ROWS * ARCc;   // 12800*500
  float* relHe = Ubuf + ROWS * ARCc;    // 12800*104 (col 100 = 1, 101..103 = 0)
  float* relDo = relHe + ROWS * RELP;   // 12800*100
  float* Tbuf  = relDo + ROWS * RELc;   // 12800*104 (pad cols stay 0)

  // 1) embeddings, time-major (t*B+b, 200)
  embed_kernel<<<(ROWS * 200 + 255) / 256, 256, 0, stream>>>(words, feats, wemb, femb, xA);
  // zero the padded rel buffers once so pad columns contribute 0 to K=104 GEMMs
  zero_kernel<<<(ROWS * RELP + 255) / 256, 256, 0, stream>>>(relHe, ROWS * RELP);
  zero_kernel<<<(ROWS * RELP + 255) / 256, 256, 0, stream>>>(Tbuf, ROWS * RELP);

  // 2) 3-layer BiLSTM
  float* xbufs[2] = { xA, xB };
  int cur = 0, din = 200;
  for (int l = 0; l < 3; ++l) {
    float* xin  = xbufs[cur];
    float* xout = xbufs[cur ^ 1];
    for (int d = 0; d < 2; ++d) {
      // XW = x @ Wih^T + bih + bhh   (hoisted out of the scan)
      run_gemm(stream, GemmArgs{ xin, 0, din,  Wih[l][d], 0, din,
                                 nullptr, 0, 0,  XW, 0, GDc, 1,
                                 bih[l][d], 0, bhh[l][d],  ROWS, GDc, din, 1, 0 }, 1);
      zero_kernel<<<(2 * Bc * HIDc + 255) / 256, 256, 0, stream>>>(hbuf, 2 * Bc * HIDc);
      for (int s = 0; s < Lc; ++s) {
        int t = d ? (Lc - 1 - s) : s;
        // g = XW[t] + h @ Whh^T
        run_gemm(stream, GemmArgs{ hbuf, 0, HIDc,  Whh[l][d], 0, HIDc,
                                   XW + (long long)t * Bc * GDc, 0, GDc,
                                   gbuf, 0, GDc, 1,
                                   nullptr, 0, nullptr,  Bc, GDc, HIDc, 1, 0 }, 1);
        lstm_cell_kernel<<<(Bc * HIDc + 255) / 256, 256, 0, stream>>>(
            gbuf, words, hbuf, cbuf, xout, t, d * HIDc, 800);
      }
    }
    cur ^= 1; din = 800;
  }
  float* xfin = xbufs[cur];   // (12800, 800)

  // 3) MLPs (leaky_relu 0.1)
  run_gemm(stream, GemmArgs{ xfin, 0, 800, arcHW, 0, 800, nullptr, 0, 0,
                             arcHo, 0, ARCc, 1, arcHb, 0, nullptr,
                             ROWS, ARCc, 800, 1, 1 }, 1);
  run_gemm(stream, GemmArgs{ xfin, 0, 800, arcDW, 0, 800, nullptr, 0, 0,
                             arcDo, 0, ARCc, 1, arcDb, 0, nullptr,
                             ROWS, ARCc, 800, 1, 1 }, 1);
  run_gemm(stream, GemmArgs{ xfin, 0, 800, relHW, 0, 800, nullptr, 0, 0,
                             relHe, 0, RELP, 1, relHb, 0, nullptr,
                             ROWS, RELc, 800, 1, 1 }, 1);
  run_gemm(stream, GemmArgs{ xfin, 0, 800, relDW, 0, 800, nullptr, 0, 0,
                             relDo, 0, RELc, 1, relDb, 0, nullptr,
                             ROWS, RELc, 800, 1, 1 }, 1);
  set_ones_col<<<(ROWS + 255) / 256, 256, 0, stream>>>(relHe, ROWS, RELP, 100);

  // 4) arcs: U = arc_d @ W_arc[:500] + W_arc[500]  (the +1 column folded into bias)
  run_gemm(stream, GemmArgs{ arcDo, 0, ARCc,  Warc, 0, ARCc, nullptr, 0, 0,
                             Ubuf, 0, ARCc, 1,  Warc + ARCc * ARCc, 0, nullptr,
                             ROWS, ARCc, ARCc, 1, 0 }, 0);
  // s_arc[b,x,y] = U[b,x,:] . arc_h[b,y,:]   (batched over b; rows strided by B)
  run_gemm(stream, GemmArgs{ Ubuf, ARCc, Bc * ARCc,  arcHo, ARCc, Bc * ARCc,
                             nullptr, 0, 0,  out, Lc * Lc, Lc, 1,
                             nullptr, 0, nullptr,  Lc, Lc, ARCc, Bc, 0 }, 1);
  arc_mask_kernel<<<(Bc * Lc * Lc + 255) / 256, 256, 0, stream>>>(out, words);

  // 5) rels, per label o: T = rel_d @ W_rel[o,:100,:] + W_rel[o,100,:]
  //    s_rel[b,:,:,o] = T[b] @ rel_h_ext[b]^T   (strided write, col stride 50)
  for (int o = 0; o < LBLc; ++o) {
    const float* Wo = Wrel + (long long)o * 101 * 101;
    run_gemm(stream, GemmArgs{ relDo, 0, RELc,  Wo, 0, 101, nullptr, 0, 0,
                               Tbuf, 0, RELP, 1,  Wo + 100 * 101, 0, nullptr,
                               ROWS, 101, RELc, 1, 0 }, 0);
    run_gemm(stream, GemmArgs{ Tbuf, RELP, Bc * RELP,  relHe, RELP, Bc * RELP,
                               nullptr, 0, 0,
                               s_rel + o, (long long)Lc * Lc * LBLc, Lc * LBLc, LBLc,
                               nullptr, 0, nullptr,  Lc, Lc, RELP, Bc, 0 }, 1);
  }
}